// GraphSELayer_31860067402236
// MI455X (gfx1250) — compile-verified
//
#include <hip/hip_runtime.h>

// ---------------------------------------------------------------------------
// Graph squeeze-excite layer for MI455X (gfx1250, wave32).
//
// Bandwidth analysis: x (512MB) must stream from HBM twice (pool pass +
// scale pass) since scale depends on a global reduction and x > L2 (192MB).
// Floor = (512 read + 512 read + 512 write) MB / 23.3 TB/s ~= 66us.
// The MLP (512x256x16 GEMMs) is ~8 MFLOP -> done with V_WMMA_F32_16X16X4_F32
// in fp32 to match reference precision exactly.
// ---------------------------------------------------------------------------

typedef __attribute__((ext_vector_type(2))) float v2f;
typedef __attribute__((ext_vector_type(4))) float fv4;
typedef __attribute__((ext_vector_type(8))) float v8f;

#define GRAPHS 512
#define CH     256
#define HID    16
#define C4     (CH / 4)   // 64 float4 per row

// ---------------------------------------------------------------------------
// K0: zero sums + counts accumulators (deterministic across graph replays)
// ---------------------------------------------------------------------------
__global__ void se_zero_kernel(float* __restrict__ ws) {
  int i = blockIdx.x * blockDim.x + threadIdx.x;
  if (i < GRAPHS * CH + GRAPHS) ws[i] = 0.0f;
}

// ---------------------------------------------------------------------------
// K1: segment sum + counts. batch is sorted, so each thread run-length
// accumulates its float4 column slice and flushes to global atomics only on
// segment boundaries (~2-3 flushes per thread). x read with NT 128-bit loads.
// Thread layout: lane = tid&63 -> float4 column, q = tid>>6 -> row phase.
// A wave (32 lanes, one quad half) reads 512B contiguous per row: coalesced.
// ---------------------------------------------------------------------------
__global__ __launch_bounds__(256)
void se_segsum_kernel(const fv4* __restrict__ x4, const int* __restrict__ batch,
                      float* __restrict__ sums, float* __restrict__ counts,
                      int n, int rowsPerBlock) {
  const int lane = threadIdx.x & 63;
  const int q    = threadIdx.x >> 6;
  const int rowBase = blockIdx.x * rowsPerBlock;
  const int rowEnd  = min(rowBase + rowsPerBlock, n);

  fv4 acc = (fv4)0.0f;
  int curSeg = -1;
  int runCnt = 0;

  for (int r = rowBase + q; r < rowEnd; r += 4) {
    int s = batch[r];              // same addr across the wave -> 1 request
    if (s != curSeg) {
      if (curSeg >= 0) {
        float* dst = sums + curSeg * CH + lane * 4;
        atomicAdd(dst + 0, acc.x); atomicAdd(dst + 1, acc.y);
        atomicAdd(dst + 2, acc.z); atomicAdd(dst + 3, acc.w);
        if (lane == 0) atomicAdd(&counts[curSeg], (float)runCnt);
      }
      acc = (fv4)0.0f;
      runCnt = 0;
      curSeg = s;
    }
    fv4 xv = __builtin_nontemporal_load(&x4[(size_t)r * C4 + lane]);
    acc += xv;
    runCnt++;
  }
  if (curSeg >= 0) {
    float* dst = sums + curSeg * CH + lane * 4;
    atomicAdd(dst + 0, acc.x); atomicAdd(dst + 1, acc.y);
    atomicAdd(dst + 2, acc.z); atomicAdd(dst + 3, acc.w);
    if (lane == 0) atomicAdd(&counts[curSeg], (float)runCnt);
  }
}

// ---------------------------------------------------------------------------
// K2: SE MLP with fp32 WMMA. One wave32 per 16-graph strip (32 blocks).
//
// V_WMMA_F32_16X16X4_F32 layouts (ISA 7.12.2):
//   A 16x4:  lane L -> m = L&15, VGPR v -> k = 2*(L>>4) + v
//   B 4x16:  lane L -> n = L&15, VGPR v -> k = 2*(L>>4) + v   (mirror of A)
//   C 16x16: lane L -> N = L&15, VGPR v -> M = v + 8*(L>>4)
// GEMM1: H[16,16] = relu(mean[16,256] @ W1[256,16]), 64 K-steps.
// H restaged through LDS to convert C-layout -> A-layout for GEMM2.
// GEMM2: scale[16,256] = sigmoid(H @ W2), 16 N-tiles x 4 K-steps.
// ---------------------------------------------------------------------------
__global__ __launch_bounds__(32)
void se_mlp_kernel(const float* __restrict__ sums, const float* __restrict__ counts,
                   const float* __restrict__ W1, const float* __restrict__ W2,
                   float* __restrict__ scale) {
  __shared__ float Hs[16 * 16];

  const int lane = threadIdx.x;   // 0..31
  const int m    = lane & 15;
  const int hs   = lane >> 4;     // half-select: which K/M half this lane holds
  const int g0   = blockIdx.x * 16;
  const int g    = g0 + m;

  const float rinv = 1.0f / fmaxf(counts[g], 1.0f);

  // ---- GEMM1: mean @ W1 -> H (16x16 accumulator, only semantics of 16 cols)
  v8f acc = {};
  for (int k0 = 0; k0 < CH; k0 += 4) {
    const int ka = k0 + hs * 2;
    v2f a, b;
    a.x = sums[g * CH + ka]     * rinv;        // A[m][ka]
    a.y = sums[g * CH + ka + 1] * rinv;        // A[m][ka+1]
    b.x = W1[ka * HID + m];                    // B[ka][m]   (W1 is [256,16])
    b.y = W1[(ka + 1) * HID + m];              // B[ka+1][m]
    acc = __builtin_amdgcn_wmma_f32_16x16x4_f32(
        false, a, false, b, (short)0, acc, false, false);
  }

  // ReLU + restage C-layout -> LDS row-major H[M][N]
#pragma unroll
  for (int v = 0; v < 8; ++v)
    Hs[(v + 8 * hs) * 16 + m] = fmaxf(acc[v], 0.0f);
  __syncthreads();

  // ---- GEMM2: H @ W2 -> sigmoid -> scale[g0..g0+15][0..255]
  for (int n0 = 0; n0 < CH; n0 += 16) {
    v8f c2 = {};
#pragma unroll
    for (int k0 = 0; k0 < HID; k0 += 4) {
      const int ka = k0 + hs * 2;
      v2f a, b;
      a.x = Hs[m * 16 + ka];                   // A[m][ka]
      a.y = Hs[m * 16 + ka + 1];
      b.x = W2[ka * CH + n0 + m];              // B[ka][n0+m] (W2 is [16,256])
      b.y = W2[(ka + 1) * CH + n0 + m];
      c2 = __builtin_amdgcn_wmma_f32_16x16x4_f32(
          false, a, false, b, (short)0, c2, false, false);
    }
#pragma unroll
    for (int v = 0; v < 8; ++v) {
      const float sv = 1.0f / (1.0f + __expf(-c2[v]));
      scale[(size_t)(g0 + v + 8 * hs) * CH + n0 + m] = sv;
    }
  }
}

// ---------------------------------------------------------------------------
// K3: out = x * scale[batch]. NT 128-bit streaming for x/out; scale (512KB)
// and batch (2MB) stay resident in L2 with heavy reuse (~977 rows/graph).
// ---------------------------------------------------------------------------
__global__ __launch_bounds__(256)
void se_apply_kernel(const fv4* __restrict__ x4, const int* __restrict__ batch,
                     const fv4* __restrict__ scale4, fv4* __restrict__ out4,
                     int n, int rowsPerBlock) {
  const int lane = threadIdx.x & 63;
  const int q    = threadIdx.x >> 6;
  const int rowBase = blockIdx.x * rowsPerBlock;
  const int rowEnd  = min(rowBase + rowsPerBlock, n);

  for (int r = rowBase + q; r < rowEnd; r += 4) {
    const int s = batch[r];
    fv4 sc = scale4[(size_t)s * C4 + lane];
    fv4 xv = __builtin_nontemporal_load(&x4[(size_t)r * C4 + lane]);
    __builtin_nontemporal_store(xv * sc, &out4[(size_t)r * C4 + lane]);
  }
}

// ---------------------------------------------------------------------------
extern "C" void kernel_launch(void* const* d_in, const int* in_sizes, int n_in,
                              void* d_out, int out_size, void* d_ws, size_t ws_size,
                              hipStream_t stream) {
  const float* x     = (const float*)d_in[0];
  const int*   batch = (const int*)d_in[1];
  const float* W1    = (const float*)d_in[2];
  const float* W2    = (const float*)d_in[3];
  float*       out   = (float*)d_out;

  const int n = in_sizes[1];                 // number of nodes (500000)

  // workspace: sums | counts | scale
  float* sums   = (float*)d_ws;
  float* counts = sums + GRAPHS * CH;
  float* scale  = counts + GRAPHS;

  // K0: zero accumulators
  const int zeroTotal = GRAPHS * CH + GRAPHS;
  se_zero_kernel<<<(zeroTotal + 255) / 256, 256, 0, stream>>>(sums);

  // K1: segment sums + counts (streams x once)
  const int rows1 = 512;
  se_segsum_kernel<<<(n + rows1 - 1) / rows1, 256, 0, stream>>>(
      (const fv4*)x, batch, sums, counts, n, rows1);

  // K2: WMMA MLP -> scale[512,256]
  se_mlp_kernel<<<GRAPHS / 16, 32, 0, stream>>>(sums, counts, W1, W2, scale);

  // K3: out = x * scale[batch] (streams x + out)
  const int rows3 = 64;
  se_apply_kernel<<<(n + rows3 - 1) / rows3, 256, 0, stream>>>(
      (const fv4*)x, batch, (const fv4*)scale, (fv4*)out, n, rows3);
}